// GATLayer_18734647345426
// MI455X (gfx1250) — compile-verified
//
#include <hip/hip_runtime.h>
#include <hip/hip_bf16.h>

typedef __attribute__((ext_vector_type(2))) float v2f;
typedef __attribute__((ext_vector_type(8))) float v8f;

#define B_      4
#define N_      1024
#define IN_DIM  128
#define OUT_DIM 128
#define HEADS   8
#define DH      16   // OUT_DIM / HEADS

// -------------------------------------------------------------------------
// Kernel 1: xt = x @ W   (fp32 WMMA 16x16x4)
// x: (B*N, 128) row-major, W: (128, 128) row-major, xt: (B*N, 128)
// One wave computes one 16x16 output tile; 32 WMMAs along K.
// A layout (16x4 f32): lane<16 holds M=lane, elems {K0,K1}; lane>=16: {K2,K3}
// B layout (4x16 f32): lane%16 = N, elem v corresponds K = v + 2*(lane>=16)
// C/D layout: lane%16 = N, vgpr r -> row M = r + 8*(lane>=16)
// -------------------------------------------------------------------------
__global__ __launch_bounds__(32)
void gat_xw_kernel(const float* __restrict__ x,
                   const float* __restrict__ W,
                   float* __restrict__ xt)
{
    const int lane = threadIdx.x;
    const int half = lane >> 4;        // 0 or 1
    const int ln   = lane & 15;

    const int row_base = blockIdx.x * 16;   // over B*N
    const int col_base = blockIdx.y * 16;   // over OUT_DIM

    const float* arow = x + (size_t)(row_base + ln) * IN_DIM + 2 * half;
    const float* bcol = W + (size_t)(2 * half) * OUT_DIM + col_base + ln;

    v8f acc = {};
    for (int k0 = 0; k0 < IN_DIM; k0 += 4) {
        v2f a, b;
        a.x = arow[k0];
        a.y = arow[k0 + 1];
        b.x = bcol[(size_t)k0 * OUT_DIM];
        b.y = bcol[(size_t)(k0 + 1) * OUT_DIM];
        acc = __builtin_amdgcn_wmma_f32_16x16x4_f32(
            /*neg_a=*/false, a, /*neg_b=*/false, b,
            /*c_mod=*/(short)0, acc, /*reuse_a=*/false, /*reuse_b=*/false);
    }

#pragma unroll
    for (int r = 0; r < 8; ++r) {
        const int row = row_base + r + 8 * half;
        xt[(size_t)row * OUT_DIM + col_base + ln] = acc[r];
    }
}

// -------------------------------------------------------------------------
// Kernel 2: src[b,n,h] = xt[b,n,h,:16] . a[:16] ; dst uses a[16:32]
// One thread per (b,n,h).
// -------------------------------------------------------------------------
__global__ __launch_bounds__(256)
void gat_srcdst_kernel(const float* __restrict__ xt,
                       const float* __restrict__ a,
                       float* __restrict__ srcv,
                       float* __restrict__ dstv)
{
    const int idx = blockIdx.x * blockDim.x + threadIdx.x;  // over B*N*HEADS
    if (idx >= B_ * N_ * HEADS) return;
    const int row = idx / HEADS;   // b*N + n
    const int h   = idx % HEADS;

    const float* v = xt + (size_t)row * OUT_DIM + h * DH;
    float s = 0.f, d = 0.f;
#pragma unroll
    for (int t = 0; t < DH; ++t) {
        const float xv = v[t];
        s += xv * a[t];
        d += xv * a[DH + t];
    }
    srcv[(size_t)row * HEADS + h] = s;
    dstv[(size_t)row * HEADS + h] = d;
}

// -------------------------------------------------------------------------
// Kernel 3: flash-style masked attention + PV via fp32 WMMA.
// Block = 1 wave, handles (b, h, 16-row i-tile); loops over 64 j-tiles.
//   e[i,j]   = leaky_relu(src_i + dst_j, 0.2), masked by (adj|eye)
//   alpha    = softmax_j(e)  (online max/sum)
//   O[i,:16] = alpha @ xt[b,:,h,:16]      -> relu -> out
// P entries are produced directly in the WMMA A-operand layout.
// -------------------------------------------------------------------------
__global__ __launch_bounds__(32)
void gat_attn_kernel(const float* __restrict__ xt,
                     const float* __restrict__ srcv,
                     const float* __restrict__ dstv,
                     const float* __restrict__ adj,
                     float* __restrict__ out)
{
    __shared__ float sdst[N_];   // dst[b, :, h]  (4 KB)

    const int lane = threadIdx.x;
    const int half = lane >> 4;
    const int ln   = lane & 15;

    const int i_base = blockIdx.x * 16;
    const int h      = blockIdx.y;
    const int b      = blockIdx.z;

    // stage dst row for this (b,h) into LDS
    for (int t = lane; t < N_; t += 32)
        sdst[t] = dstv[(size_t)(b * N_ + t) * HEADS + h];
    __syncthreads();

    const int   i_glob = i_base + ln;                 // row this lane scores
    const float src_i  = srcv[(size_t)(b * N_ + i_glob) * HEADS + h];
    const float* adj_row = adj + ((size_t)b * N_ + i_glob) * N_;
    const float* vbase   = xt + (size_t)b * N_ * OUT_DIM + h * DH + ln;

    float m_run = -1e30f;   // running row max (row = ln; dup on both halves)
    float l_run = 0.f;      // running row sum
    v8f   acc   = {};       // O tile in C layout

    for (int jt = 0; jt < N_ / 16; ++jt) {
        const int j_base = jt * 16;

        // ---- scores for this lane's 8 (i,j) slots (A-operand layout) ----
        float p[4][2];
        float tmax = -1e30f;
#pragma unroll
        for (int c = 0; c < 4; ++c) {
#pragma unroll
            for (int v = 0; v < 2; ++v) {
                const int j = j_base + 4 * c + v + 2 * half;
                float e = src_i + sdst[j];
                e = (e > 0.f) ? e : 0.2f * e;                 // leaky_relu
                const bool msk = (adj_row[j] != 0.f) || (i_glob == j);
                e = msk ? e : -1e30f;
                p[c][v] = e;
                tmax = fmaxf(tmax, e);
            }
        }
        tmax = fmaxf(tmax, __shfl_xor(tmax, 16, 32));         // combine halves
        const float m_new = fmaxf(m_run, tmax);
        const float scale = __expf(m_run - m_new);            // per-row (ln)

        // ---- p = exp(e - m_new), masked -> 0 ; row partial sums ----
        float psum = 0.f;
#pragma unroll
        for (int c = 0; c < 4; ++c) {
#pragma unroll
            for (int v = 0; v < 2; ++v) {
                float e = p[c][v];
                float pv = (e <= -1e29f) ? 0.f : __expf(e - m_new);
                p[c][v] = pv;
                psum += pv;
            }
        }
        psum += __shfl_xor(psum, 16, 32);
        l_run = l_run * scale + psum;
        m_run = m_new;

        // ---- rescale accumulator rows (C layout: row = r + 8*half) ----
#pragma unroll
        for (int r = 0; r < 8; ++r) {
            const float f = __shfl(scale, r + 8 * half, 32);
            acc[r] *= f;
        }

        // ---- acc += P(16x16) @ V(16x16) as four K=4 WMMAs ----
#pragma unroll
        for (int c = 0; c < 4; ++c) {
            v2f pa, vb;
            pa.x = p[c][0];
            pa.y = p[c][1];
            const int j0 = j_base + 4 * c + 2 * half;
            vb.x = vbase[(size_t)j0 * OUT_DIM];
            vb.y = vbase[(size_t)(j0 + 1) * OUT_DIM];
            acc = __builtin_amdgcn_wmma_f32_16x16x4_f32(
                false, pa, false, vb, (short)0, acc, false, false);
        }
    }

    // ---- normalize, relu, store ----
#pragma unroll
    for (int r = 0; r < 8; ++r) {
        const int row = r + 8 * half;
        const float li = __shfl(l_run, row, 32);
        float o = acc[r] / li;
        o = fmaxf(o, 0.f);
        out[(size_t)(b * N_ + i_base + row) * OUT_DIM + h * DH + ln] = o;
    }
}

// -------------------------------------------------------------------------
extern "C" void kernel_launch(void* const* d_in, const int* in_sizes, int n_in,
                              void* d_out, int out_size, void* d_ws, size_t ws_size,
                              hipStream_t stream)
{
    const float* x   = (const float*)d_in[0];  // (B,N,128)
    const float* adj = (const float*)d_in[1];  // (B,N,N)
    const float* W   = (const float*)d_in[2];  // (128,128)
    const float* a   = (const float*)d_in[3];  // (32,1)
    float* out = (float*)d_out;                // (B,N,128)

    float* xt   = (float*)d_ws;                      // B*N*128  = 2 MB
    float* srcv = xt   + (size_t)B_ * N_ * OUT_DIM;  // B*N*H    = 128 KB
    float* dstv = srcv + (size_t)B_ * N_ * HEADS;    // B*N*H    = 128 KB

    gat_xw_kernel<<<dim3(B_ * N_ / 16, OUT_DIM / 16), 32, 0, stream>>>(x, W, xt);

    gat_srcdst_kernel<<<(B_ * N_ * HEADS + 255) / 256, 256, 0, stream>>>(
        xt, a, srcv, dstv);

    gat_attn_kernel<<<dim3(N_ / 16, HEADS, B_), 32, 0, stream>>>(
        xt, srcv, dstv, adj, out);
}